// H2GCNBranch_15633680958306
// MI455X (gfx1250) — compile-verified
//
#include <hip/hip_runtime.h>
#include <hip/hip_bf16.h>

#define IN_DIM 256
#define HID    64
#define OUTW   192   // 3*HID, output row stride

typedef __attribute__((ext_vector_type(2))) float v2f;
typedef __attribute__((ext_vector_type(8))) float v8f;

// ---------------------------------------------------------------------------
// Zero the output buffer (atomics accumulate into it, harness poisons it).
// ---------------------------------------------------------------------------
__global__ void h2gcn_zero_kernel(float* __restrict__ out, long n) {
    long i = ((long)blockIdx.x * blockDim.x + threadIdx.x) * 4;
    if (i + 3 < n) {
        float4 z = make_float4(0.f, 0.f, 0.f, 0.f);
        *(float4*)(out + i) = z;
    } else {
        for (long j = i; j < n; ++j) out[j] = 0.f;
    }
}

// ---------------------------------------------------------------------------
// h0 = x @ W using V_WMMA_F32_16X16X4_F32 (fp32 WMMA, no precision loss).
// Block = 256 threads = 8 waves; each wave owns a 16-row strip and all 64
// output columns (4 x 16x16 f32 accumulators). W is staged in LDS,
// interleaved as pairs along K so B fragments are single ds_load_b64's.
//
// A-frag layout (16x4 f32, ISA 7.12.2): lane = m + 16*(k>=2), vgpr = k&1
//   -> lane L holds A[m][2*(L>>4) + {0,1}]  (contiguous K pair -> float2 load)
// B-frag layout (4x16 f32): lane = n + 16*(k>=2), vgpr = k&1
//   -> lane L holds B[2*(L>>4)+{0,1}][n]
// C/D layout: vgpr r, lane L -> M = r + 8*(L>>4), N = L&15
// ---------------------------------------------------------------------------
__global__ void __launch_bounds__(256)
h2gcn_gemm_wmma_kernel(const float* __restrict__ x,
                       const float* __restrict__ W,
                       float* __restrict__ out,
                       int nrows) {
    // W interleaved: lds[((k>>1)*HID + n)*2 + (k&1)] = W[k*HID + n]
    __shared__ float Wlds[IN_DIM * HID];

    for (int i = threadIdx.x; i < IN_DIM * HID; i += blockDim.x) {
        int k = i / HID;
        int n = i - k * HID;
        Wlds[(((k >> 1) * HID) + n) * 2 + (k & 1)] = W[i];
    }
    __syncthreads();

    const int wave = threadIdx.x >> 5;       // 0..7
    const int lane = threadIdx.x & 31;
    const int m    = lane & 15;              // row within tile / col within tile
    const int kh   = lane >> 4;              // 0 or 1 (K-half selector)

    const int rowBase = blockIdx.x * 128 + wave * 16;
    if (rowBase >= nrows) return;            // wave-uniform: EXEC stays all-1s

    v8f acc0 = {}, acc1 = {}, acc2 = {}, acc3 = {};

    const float* xrow = x + (size_t)(rowBase + m) * IN_DIM;

    for (int k = 0; k < IN_DIM; k += 4) {
        // A fragment: two consecutive K values for this lane's half
        v2f a = *(const v2f*)(xrow + k + 2 * kh);

        const int p = (k + 2 * kh) >> 1;     // interleaved pair index
        const float* bbase = &Wlds[(size_t)p * HID * 2];

        v2f b0 = *(const v2f*)(bbase + (0 * 16 + m) * 2);
        v2f b1 = *(const v2f*)(bbase + (1 * 16 + m) * 2);
        v2f b2 = *(const v2f*)(bbase + (2 * 16 + m) * 2);
        v2f b3 = *(const v2f*)(bbase + (3 * 16 + m) * 2);

        acc0 = __builtin_amdgcn_wmma_f32_16x16x4_f32(false, a, false, b0,
                                                     (short)0, acc0, false, false);
        acc1 = __builtin_amdgcn_wmma_f32_16x16x4_f32(false, a, false, b1,
                                                     (short)0, acc1, false, false);
        acc2 = __builtin_amdgcn_wmma_f32_16x16x4_f32(false, a, false, b2,
                                                     (short)0, acc2, false, false);
        acc3 = __builtin_amdgcn_wmma_f32_16x16x4_f32(false, a, false, b3,
                                                     (short)0, acc3, false, false);
    }

    // Store D: out row stride is OUTW (h0 occupies columns 0..63)
    float* obase = out + (size_t)(rowBase + 8 * kh) * OUTW + m;
#pragma unroll
    for (int r = 0; r < 8; ++r) {
        float* orow = obase + (size_t)r * OUTW;
        orow[0 * 16] = acc0[r];
        orow[1 * 16] = acc1[r];
        orow[2 * 16] = acc2[r];
        orow[3 * 16] = acc3[r];
    }
}

// ---------------------------------------------------------------------------
// SpMM scatter: out[dst[e], colOff + :64] += vals[e] * h0[src[e], :]
// 16 lanes per edge, each lane handles a float4 chunk of the 64-wide row.
// h0 is read from d_out columns 0..63 (row stride OUTW) — 25.6 MB, L2
// resident on MI455X.  Atomics are native global_atomic_add_f32.
// ---------------------------------------------------------------------------
__global__ void __launch_bounds__(256)
h2gcn_spmm_kernel(const float* __restrict__ vals,
                  const int*  __restrict__ src,
                  const int*  __restrict__ dst,
                  const float* __restrict__ h0,   // = d_out (cols 0..63)
                  float* __restrict__ out,
                  int E, int colOff) {
    long tid = (long)blockIdx.x * blockDim.x + threadIdx.x;
    int e   = (int)(tid >> 4);
    int sub = (int)(tid & 15);
    if (e >= E) return;

    float v = vals[e];
    int s = src[e];
    int d = dst[e];

    const float4 h = *(const float4*)(h0 + (size_t)s * OUTW + sub * 4);

    float* op = out + (size_t)d * OUTW + colOff + sub * 4;
    unsafeAtomicAdd(op + 0, v * h.x);
    unsafeAtomicAdd(op + 1, v * h.y);
    unsafeAtomicAdd(op + 2, v * h.z);
    unsafeAtomicAdd(op + 3, v * h.w);
}

// ---------------------------------------------------------------------------
// Launch
// ---------------------------------------------------------------------------
extern "C" void kernel_launch(void* const* d_in, const int* in_sizes, int n_in,
                              void* d_out, int out_size, void* d_ws, size_t ws_size,
                              hipStream_t stream) {
    const float* x     = (const float*)d_in[0];
    const float* W     = (const float*)d_in[1];
    const float* vals1 = (const float*)d_in[2];
    const float* vals2 = (const float*)d_in[3];
    const int*   src1  = (const int*)d_in[4];
    const int*   dst1  = (const int*)d_in[5];
    const int*   src2  = (const int*)d_in[6];
    const int*   dst2  = (const int*)d_in[7];

    const int N  = in_sizes[0] / IN_DIM;   // 100000
    const int E1 = in_sizes[2];            // 1.6M
    const int E2 = in_sizes[3];            // 3.2M

    float* out = (float*)d_out;

    // 1) zero output (atomic accumulation targets)
    {
        long n = (long)out_size;
        long n4 = (n + 3) / 4;
        int blocks = (int)((n4 + 255) / 256);
        h2gcn_zero_kernel<<<blocks, 256, 0, stream>>>(out, n);
    }

    // 2) h0 = x @ W  -> out columns [0, 64)
    {
        int blocks = (N + 127) / 128;      // 128 rows per block (8 waves x 16)
        h2gcn_gemm_wmma_kernel<<<blocks, 256, 0, stream>>>(x, W, out, N);
    }

    // 3) h1 scatter -> out columns [64, 128)
    {
        long threads = (long)E1 * 16;
        int blocks = (int)((threads + 255) / 256);
        h2gcn_spmm_kernel<<<blocks, 256, 0, stream>>>(vals1, src1, dst1,
                                                      out, out, E1, HID);
    }

    // 4) h2 scatter -> out columns [128, 192)
    {
        long threads = (long)E2 * 16;
        int blocks = (int)((threads + 255) / 256);
        h2gcn_spmm_kernel<<<blocks, 256, 0, stream>>>(vals2, src2, dst2,
                                                      out, out, E2, 2 * HID);
    }
}